// QMeasureDensity_10239202034197
// MI455X (gfx1250) — compile-verified
//
#include <hip/hip_runtime.h>

typedef __attribute__((ext_vector_type(2))) float v2f;
typedef __attribute__((ext_vector_type(8))) float v8f;

#define DIM   256          // feature dimension
#define ROWS  16           // batch rows per block
#define WAVES 16           // waves per block; wave w owns N-columns [16w,16w+16)
#define LDX   (DIM + 4)    // padded LDS row pitch (260 floats): keeps 16B row
                           // alignment and makes the strided A-fragment read
                           // (stride 260 -> bank step 4) hit all 64 banks.

__global__ __launch_bounds__(512, 1) void qmeasure_density_kernel(
    const float* __restrict__ X,    // [BATCH, 256] row-major
    const float* __restrict__ rho,  // [256, 256]  row-major
    float* __restrict__ out) {      // [BATCH]
  __shared__ float xs[ROWS * LDX];       // 16.25 KB staged X chunk
  __shared__ float qpart[WAVES * ROWS];  // per-wave partial x^T rho x

  const int tid = threadIdx.x;
  const int bid = blockIdx.x;

  // ---- stage 16x256 X chunk into LDS (float4, coalesced) ----
  const float4* src = (const float4*)(X + (size_t)bid * ROWS * DIM);
#pragma unroll
  for (int i = 0; i < 2; ++i) {
    int g4  = tid + i * 512;       // 1024 float4 total
    int row = g4 >> 6;             // 64 float4 per row
    int c4  = g4 & 63;
    *(float4*)(xs + row * LDX + c4 * 4) = src[g4];
  }
  __syncthreads();

  const int wave = tid >> 5;
  const int lane = tid & 31;
  const int hi   = lane >> 4;      // half-wave select
  const int lrow = lane & 15;      // A row / B,C column within tile
  const int koff = hi * 2;         // ISA A/B layout: hi half holds K+2,K+3
  const int ncol = wave * 16 + lrow;

  const float* ap = xs + lrow * LDX + koff;              // A from LDS
  const float* bp = rho + (size_t)koff * DIM + ncol;     // B from global (L2)

  // ---- Y tile = X_chunk(16x256) x rho(:,16w:16w+16) via fp32 WMMA ----
  v8f acc = {};
#pragma unroll 4
  for (int k = 0; k < DIM; k += 4) {
    v2f a = *(const v2f*)(ap + k);                 // ds_load_b64, no conflicts
    v2f b;
    b.x = bp[(size_t)k * DIM];                     // rho[k+koff  ][ncol]
    b.y = bp[(size_t)k * DIM + DIM];               // rho[k+koff+1][ncol]
    acc = __builtin_amdgcn_wmma_f32_16x16x4_f32(
        /*neg_a=*/false, a, /*neg_b=*/false, b,
        /*c_mod=*/(short)0, acc, /*reuse_a=*/false, /*reuse_b=*/false);
  }

  // ---- q partials: p[j] = Y[row,ncol] * X[row,ncol], row = j + 8*hi ----
  float p[8];
#pragma unroll
  for (int j = 0; j < 8; ++j) {
    int row = j + 8 * hi;
    p[j] = acc[j] * xs[row * LDX + ncol];
  }
  // butterfly-reduce over the 16 lanes of each half-wave (deterministic)
#pragma unroll
  for (int m = 1; m < 16; m <<= 1) {
#pragma unroll
    for (int j = 0; j < 8; ++j) p[j] += __shfl_xor(p[j], m, 16);
  }
  if (lrow == 0) {
#pragma unroll
    for (int j = 0; j < 8; ++j) qpart[wave * ROWS + j + 8 * hi] = p[j];
  }
  __syncthreads();

  // ---- final: out[b] = ||x||^2 * (x^T rho x), fixed-order sums ----
  if (tid < ROWS) {
    const int row = tid;
    float q = 0.f;
#pragma unroll
    for (int w = 0; w < WAVES; ++w) q += qpart[w * ROWS + row];
    float n2 = 0.f;
    const float* xr = xs + row * LDX;
#pragma unroll 8
    for (int i = 0; i < DIM; ++i) n2 += xr[i] * xr[i];
    out[(size_t)bid * ROWS + row] = n2 * q;
  }
}

extern "C" void kernel_launch(void* const* d_in, const int* in_sizes, int n_in,
                              void* d_out, int out_size, void* d_ws, size_t ws_size,
                              hipStream_t stream) {
  (void)n_in; (void)out_size; (void)d_ws; (void)ws_size;
  const float* X   = (const float*)d_in[0];   // [BATCH, 256] f32
  const float* rho = (const float*)d_in[1];   // [256, 256]  f32
  float* out = (float*)d_out;                 // [BATCH]     f32

  const int batch  = in_sizes[0] / DIM;       // 4096
  const int blocks = batch / ROWS;            // 256 blocks x 512 threads
  hipLaunchKernelGGL(qmeasure_density_kernel, dim3(blocks), dim3(512), 0,
                     stream, X, rho, out);
}